// GraphNet_89481348645121
// MI455X (gfx1250) — compile-verified
//
#include <hip/hip_runtime.h>

// ---------------------------------------------------------------------------
// GraphSAGE (3x SAGEConv + mean-pool + MLP head) for MI455X (gfx1250).
// fp32 end-to-end; GEMMs via V_WMMA_F32_16X16X4_F32 (wave32).
// ---------------------------------------------------------------------------

typedef __attribute__((ext_vector_type(2))) float v2f;
typedef __attribute__((ext_vector_type(8))) float v8f;

#define NN 100000      // N_NODES
#define NE 3200000     // N_EDGES
#define NF 128         // N_FEAT
#define HID 256
#define NG 256         // N_GRAPHS

// ---------------- degree ----------------------------------------------------
__global__ void count_deg(const int* __restrict__ dst, float* __restrict__ deg,
                          int nEdges) {
  int e = blockIdx.x * blockDim.x + threadIdx.x;
  if (e < nEdges) atomicAdd(&deg[dst[e]], 1.0f);
}

__global__ void invert_deg(float* __restrict__ deg, int n) {
  int i = blockIdx.x * blockDim.x + threadIdx.x;
  if (i < n) deg[i] = 1.0f / fmaxf(deg[i], 1.0f);
}

// ---------------- edge scatter-add (mean aggregation numerator) -------------
// flat (edge, feature) indexing; F is a power of two (128 or 256) -> shift.
__global__ void scatter_feat(const float* __restrict__ h,
                             const int* __restrict__ src,
                             const int* __restrict__ dst,
                             float* __restrict__ agg,
                             int nEdges, int shift) {
  long long gid = (long long)blockIdx.x * blockDim.x + threadIdx.x;
  int e = (int)(gid >> shift);
  if (e >= nEdges) return;
  int f = (int)(gid & ((1 << shift) - 1));
  int s = src[e];
  int d = dst[e];
  atomicAdd(&agg[((long long)d << shift) + f], h[((long long)s << shift) + f]);
}

// ---------------- fused SAGE layer GEMM -------------------------------------
// hout = ReLU( (agg .* invdeg) @ Wl^T + bl + hin @ Wr^T ),  out dim = HID.
// Block: 256 threads = 8 waves; tile 64 rows x 32 cols; each wave one 16x16
// WMMA tile; K staged in LDS in chunks of 64.
template <int DIN>
__global__ __launch_bounds__(256) void sage_wmma(
    const float* __restrict__ hin, const float* __restrict__ agg,
    const float* __restrict__ invdeg,
    const float* __restrict__ Wl, const float* __restrict__ bl,
    const float* __restrict__ Wr,
    float* __restrict__ hout, int nNodes) {
  constexpr int KC = 64;
  constexpr int LDK = KC + 4;  // pad to avoid LDS bank conflicts

  __shared__ float sMean[64 * LDK];
  __shared__ float sH[64 * LDK];
  __shared__ float sWl[32 * LDK];
  __shared__ float sWr[32 * LDK];

  const int tid = threadIdx.x;
  const int lane = tid & 31;
  const int wave = tid >> 5;   // 0..7
  const int wm = wave & 3;     // wave row within tile (4)
  const int wn = wave >> 2;    // wave col within tile (2)
  const int half = lane >> 4;  // 0: K0/K1 lanes, 1: K2/K3 lanes
  const int l16 = lane & 15;

  const int rowBase = blockIdx.x * 64;
  const int colBase = blockIdx.y * 32;

  v8f c = {};

  for (int k0 = 0; k0 < DIN; k0 += KC) {
    // stage A panels: 64 x 64 elements, 16 per thread, coalesced along k
#pragma unroll
    for (int i = 0; i < 16; ++i) {
      int idx = tid + i * 256;
      int r = idx >> 6;
      int k = idx & 63;
      int row = rowBase + r;
      float m = 0.0f, hv = 0.0f;
      if (row < nNodes) {
        long long off = (long long)row * DIN + k0 + k;
        m = agg[off] * invdeg[row];
        hv = hin[off];
      }
      sMean[r * LDK + k] = m;
      sH[r * LDK + k] = hv;
    }
    // stage weight panels: 32 x 64 elements, 8 per thread
#pragma unroll
    for (int i = 0; i < 8; ++i) {
      int idx = tid + i * 256;
      int n = idx >> 6;
      int k = idx & 63;
      long long off = (long long)(colBase + n) * DIN + k0 + k;
      sWl[n * LDK + k] = Wl[off];
      sWr[n * LDK + k] = Wr[off];
    }
    __syncthreads();

    const float* aMrow = &sMean[(wm * 16 + l16) * LDK];
    const float* aHrow = &sH[(wm * 16 + l16) * LDK];
    const float* bLrow = &sWl[(wn * 16 + l16) * LDK];
    const float* bRrow = &sWr[(wn * 16 + l16) * LDK];

    // A 16x4 f32 layout: lanes 0-15 -> M=lane, {K0,K1}; lanes 16-31 -> {K2,K3}
    // B 4x16 f32 layout: lanes 0-15 -> N=lane, {K0,K1}; lanes 16-31 -> {K2,K3}
#pragma unroll
    for (int k = 0; k < KC; k += 4) {
      int kk = k + half * 2;
      v2f am; am.x = aMrow[kk]; am.y = aMrow[kk + 1];
      v2f bw; bw.x = bLrow[kk]; bw.y = bLrow[kk + 1];
      c = __builtin_amdgcn_wmma_f32_16x16x4_f32(false, am, false, bw,
                                                (short)0, c, false, false);
      v2f ah; ah.x = aHrow[kk]; ah.y = aHrow[kk + 1];
      v2f br; br.x = bRrow[kk]; br.y = bRrow[kk + 1];
      c = __builtin_amdgcn_wmma_f32_16x16x4_f32(false, ah, false, br,
                                                (short)0, c, false, false);
    }
    __syncthreads();
  }

  // C/D layout: VGPR r -> M = r (lanes 0-15) or r+8 (lanes 16-31), N = l16
  const int col = colBase + wn * 16 + l16;
  const float bias = bl[col];
  const int rbase = rowBase + wm * 16 + half * 8;
#pragma unroll
  for (int r = 0; r < 8; ++r) {
    int row = rbase + r;
    if (row < nNodes) {
      float v = c[r] + bias;
      hout[(long long)row * HID + col] = v > 0.0f ? v : 0.0f;
    }
  }
}

// ---------------- global mean pool ------------------------------------------
__global__ void pool_sum(const float* __restrict__ h,
                         const int* __restrict__ batch,
                         float* __restrict__ gsum, int nNodes) {
  long long gid = (long long)blockIdx.x * blockDim.x + threadIdx.x;
  int n = (int)(gid >> 8);
  if (n >= nNodes) return;
  int f = (int)(gid & 255);
  atomicAdd(&gsum[batch[n] * HID + f], h[((long long)n << 8) + f]);
}

__global__ void pool_cnt(const int* __restrict__ batch,
                         float* __restrict__ gcnt, int nNodes) {
  int n = blockIdx.x * blockDim.x + threadIdx.x;
  if (n < nNodes) atomicAdd(&gcnt[batch[n]], 1.0f);
}

__global__ void pool_norm(float* __restrict__ gsum,
                          const float* __restrict__ gcnt) {
  int gid = blockIdx.x * blockDim.x + threadIdx.x;  // NG*HID threads
  int g = gid >> 8;
  gsum[gid] = gsum[gid] / fmaxf(gcnt[g], 1.0f);
}

// ---------------- FC head ---------------------------------------------------
__global__ void fc_layer(const float* __restrict__ in, const float* __restrict__ W,
                         const float* __restrict__ b, float* __restrict__ out,
                         int G, int Din, int Dout, int relu) {
  int gid = blockIdx.x * blockDim.x + threadIdx.x;
  if (gid >= G * Dout) return;
  int g = gid / Dout;
  int o = gid - g * Dout;
  float s = b[o];
  const float* ip = in + (long long)g * Din;
  const float* wp = W + (long long)o * Din;
  for (int k = 0; k < Din; ++k) s += ip[k] * wp[k];
  if (relu) s = fmaxf(s, 0.0f);
  out[gid] = s;
}

// ---------------------------------------------------------------------------
extern "C" void kernel_launch(void* const* d_in, const int* in_sizes, int n_in,
                              void* d_out, int out_size, void* d_ws,
                              size_t ws_size, hipStream_t stream) {
  const float* x = (const float*)d_in[0];
  const int* ei = (const int*)d_in[1];
  const int* batch = (const int*)d_in[2];
  const float* Wl0 = (const float*)d_in[3];
  const float* bl0 = (const float*)d_in[4];
  const float* Wr0 = (const float*)d_in[5];
  const float* Wl1 = (const float*)d_in[6];
  const float* bl1 = (const float*)d_in[7];
  const float* Wr1 = (const float*)d_in[8];
  const float* Wl2 = (const float*)d_in[9];
  const float* bl2 = (const float*)d_in[10];
  const float* Wr2 = (const float*)d_in[11];
  const float* Wf0 = (const float*)d_in[12];
  const float* bf0 = (const float*)d_in[13];
  const float* Wf1 = (const float*)d_in[14];
  const float* bf1 = (const float*)d_in[15];
  const float* Wf2 = (const float*)d_in[16];
  const float* bf2 = (const float*)d_in[17];

  const int* src = ei;
  const int* dst = ei + NE;

  // workspace layout (floats)
  float* bufA = (float*)d_ws;                       // NN*HID
  float* bufB = bufA + (size_t)NN * HID;            // NN*HID
  float* agg = bufB + (size_t)NN * HID;             // NN*HID
  float* invd = agg + (size_t)NN * HID;             // NN
  float* gsum = invd + NN;                          // NG*HID
  float* gcnt = gsum + (size_t)NG * HID;            // NG
  float* fc1 = gcnt + NG;                           // NG*128
  float* fc2 = fc1 + (size_t)NG * 128;              // NG*64

  const dim3 gemmGrid((NN + 63) / 64, HID / 32);

  // degrees -> inverse counts
  (void)hipMemsetAsync(invd, 0, (size_t)NN * sizeof(float), stream);
  count_deg<<<(NE + 255) / 256, 256, 0, stream>>>(dst, invd, NE);
  invert_deg<<<(NN + 255) / 256, 256, 0, stream>>>(invd, NN);

  // ---- layer 0: x(128) -> bufA(256) ----
  (void)hipMemsetAsync(agg, 0, (size_t)NN * NF * sizeof(float), stream);
  {
    long long work = (long long)NE * NF;
    scatter_feat<<<(unsigned)((work + 255) / 256), 256, 0, stream>>>(
        x, src, dst, agg, NE, 7);
  }
  sage_wmma<NF><<<gemmGrid, 256, 0, stream>>>(x, agg, invd, Wl0, bl0, Wr0,
                                              bufA, NN);

  // ---- layer 1: bufA -> bufB ----
  (void)hipMemsetAsync(agg, 0, (size_t)NN * HID * sizeof(float), stream);
  {
    long long work = (long long)NE * HID;
    scatter_feat<<<(unsigned)((work + 255) / 256), 256, 0, stream>>>(
        bufA, src, dst, agg, NE, 8);
  }
  sage_wmma<HID><<<gemmGrid, 256, 0, stream>>>(bufA, agg, invd, Wl1, bl1, Wr1,
                                               bufB, NN);

  // ---- layer 2: bufB -> bufA ----
  (void)hipMemsetAsync(agg, 0, (size_t)NN * HID * sizeof(float), stream);
  {
    long long work = (long long)NE * HID;
    scatter_feat<<<(unsigned)((work + 255) / 256), 256, 0, stream>>>(
        bufB, src, dst, agg, NE, 8);
  }
  sage_wmma<HID><<<gemmGrid, 256, 0, stream>>>(bufB, agg, invd, Wl2, bl2, Wr2,
                                               bufA, NN);

  // ---- global mean pool ----
  (void)hipMemsetAsync(gsum, 0, (size_t)NG * HID * sizeof(float), stream);
  (void)hipMemsetAsync(gcnt, 0, (size_t)NG * sizeof(float), stream);
  {
    long long work = (long long)NN * HID;
    pool_sum<<<(unsigned)((work + 255) / 256), 256, 0, stream>>>(bufA, batch,
                                                                 gsum, NN);
  }
  pool_cnt<<<(NN + 255) / 256, 256, 0, stream>>>(batch, gcnt, NN);
  pool_norm<<<(NG * HID) / 256, 256, 0, stream>>>(gsum, gcnt);

  // ---- FC head: 256 -> 128 -> 64 -> 1 ----
  fc_layer<<<(NG * 128 + 255) / 256, 256, 0, stream>>>(gsum, Wf0, bf0, fc1, NG,
                                                       HID, 128, 1);
  fc_layer<<<(NG * 64 + 255) / 256, 256, 0, stream>>>(fc1, Wf1, bf1, fc2, NG,
                                                      128, 64, 1);
  fc_layer<<<(NG + 255) / 256, 256, 0, stream>>>(fc2, Wf2, bf2, (float*)d_out,
                                                 NG, 64, 1, 0);
}